// GCN_14516989460622
// MI455X (gfx1250) — compile-verified
//
#include <hip/hip_runtime.h>

#define N_NODES 100000
#define N_EDGES 1600000
#define IN_F 128
#define H_F 64

typedef float v2f __attribute__((ext_vector_type(2)));
typedef float v8f __attribute__((ext_vector_type(8)));

// ---------------------------------------------------------------- degree pass
__global__ __launch_bounds__(256) void k_init_deg(float* __restrict__ deg) {
    int i = blockIdx.x * 256 + threadIdx.x;
    if (i < N_NODES) deg[i] = 1.0f;                  // self-loop contribution
}

__global__ __launch_bounds__(256) void k_edge_deg(const int* __restrict__ dst,
                                                  float* __restrict__ deg) {
    int e = blockIdx.x * 256 + threadIdx.x;
    if (e < N_EDGES) atomicAdd(&deg[dst[e]], 1.0f);
}

__global__ __launch_bounds__(256) void k_dinv(const float* __restrict__ deg,
                                              float* __restrict__ dinv) {
    int i = blockIdx.x * 256 + threadIdx.x;
    if (i < N_NODES) {
        float d = deg[i];
        dinv[i] = d > 0.0f ? rsqrtf(d) : 0.0f;
    }
}

// ------------------------------------------------- GEMM1: h = x @ W1 via WMMA
// Each wave computes a 16x64 strip of h. W1 (128x64 f32 = 32KB) staged in LDS
// in a K-pair-interleaved layout so every B fragment (K,K+1 at one N) is a
// contiguous 8-byte pair -> single ds_load_b64 directly into an even-aligned
// VGPR pair (no v_mov gather before the WMMA).
// f32 WMMA (16x16x4) keeps exact fp32 semantics vs the reference.
__global__ __launch_bounds__(256) void k_gemm1(const float* __restrict__ x,
                                               const float* __restrict__ W1,
                                               float* __restrict__ h) {
    __shared__ float sW[IN_F * H_F];
    // swizzled stage: sW[(k>>1)*128 + n*2 + (k&1)] = W1[k*64 + n]
    for (int i = threadIdx.x; i < IN_F * H_F; i += 256) {
        const int k = i >> 6, n = i & 63;
        sW[((k >> 1) << 7) + (n << 1) + (k & 1)] = W1[i];
    }
    __syncthreads();

    const int lane    = threadIdx.x & 31;
    const int wave    = threadIdx.x >> 5;
    const int rowBase = (blockIdx.x * 8 + wave) * 16;
    if (rowBase >= N_NODES) return;                  // wave-uniform: EXEC stays all-1s

    const int half = lane >> 4;                      // 0: lanes 0-15, 1: lanes 16-31
    const int l15  = lane & 15;
    const float* xrow = x + (size_t)(rowBase + l15) * IN_F;

    v8f acc[4] = {};                                 // 4 N-tiles of 16x16 f32

    for (int k0 = 0; k0 < IN_F; k0 += 4) {
        // A frag (16x4 f32): v0 = K(k0 | k0+2), v1 = K(k0+1 | k0+3) per half
        const int ks = k0 + half * 2;
        v2f a = *(const v2f*)(xrow + ks);
        // B frag base for this K pair; tile t at n = t*16 + l15 -> +t*32 floats
        const float* bp = &sW[((ks >> 1) << 7) + (l15 << 1)];
#pragma unroll
        for (int t = 0; t < 4; ++t) {
            v2f b = *(const v2f*)(bp + (t << 5));    // ds_load_b64
            acc[t] = __builtin_amdgcn_wmma_f32_16x16x4_f32(
                false, a, false, b, (short)0, acc[t], false, false);
        }
    }

    // D layout: VGPR r -> row (rowBase + r + 8*half), col = tile*16 + l15
#pragma unroll
    for (int t = 0; t < 4; ++t) {
        const int col = t * 16 + l15;
#pragma unroll
        for (int r = 0; r < 8; ++r) {
            const int row = rowBase + r + 8 * half;
            h[(size_t)row * H_F + col] = acc[t][r];
        }
    }
}

// ------------------------------------- layer1 aggregation: self-loop init pass
__global__ __launch_bounds__(256) void k_self1(const float* __restrict__ h,
                                               const float* __restrict__ dinv,
                                               float* __restrict__ agg) {
    int t = blockIdx.x * 256 + threadIdx.x;          // N_NODES*16 threads
    if (t >= N_NODES * 16) return;
    int node = t >> 4, fg = t & 15;
    float n = dinv[node]; n *= n;                    // self-loop norm = dinv^2
    float4 v = *((const float4*)(h + (size_t)node * H_F) + fg);
    float4 o = make_float4(v.x * n, v.y * n, v.z * n, v.w * n);
    *((float4*)(agg + (size_t)node * H_F) + fg) = o;
}

// ------------------------- layer1 aggregation: edge scatter (16 lanes / edge)
__global__ __launch_bounds__(256) void k_edge1(const int* __restrict__ src,
                                               const int* __restrict__ dst,
                                               const float* __restrict__ h,
                                               const float* __restrict__ dinv,
                                               float* __restrict__ agg) {
    int t = blockIdx.x * 256 + threadIdx.x;          // N_EDGES*16 threads
    if (t >= N_EDGES * 16) return;
    int e = t >> 4, fg = t & 15;
    int s = src[e], d = dst[e];
    float norm = dinv[s] * dinv[d];
    float4 v = *((const float4*)(h + (size_t)s * H_F) + fg);
    float* o = agg + (size_t)d * H_F + fg * 4;
    atomicAdd(o + 0, v.x * norm);
    atomicAdd(o + 1, v.y * norm);
    atomicAdd(o + 2, v.z * norm);
    atomicAdd(o + 3, v.w * norm);
}

// --------------- fused bias + ReLU + (h1 @ W2): one wave per node, K=64 dot
__global__ __launch_bounds__(256) void k_dot(const float* __restrict__ agg,
                                             const float* __restrict__ b1,
                                             const float* __restrict__ W2,
                                             float* __restrict__ z) {
    int gid  = blockIdx.x * 256 + threadIdx.x;
    int node = gid >> 5;
    int lane = threadIdx.x & 31;
    if (node >= N_NODES) return;                     // wave-uniform exit
    float2 hv = *((const float2*)(agg + (size_t)node * H_F) + lane);
    float2 bb = *((const float2*)b1 + lane);
    float2 ww = *((const float2*)W2 + lane);
    float s = fmaxf(hv.x + bb.x, 0.0f) * ww.x + fmaxf(hv.y + bb.y, 0.0f) * ww.y;
#pragma unroll
    for (int off = 16; off > 0; off >>= 1) s += __shfl_xor(s, off, 32);
    if (lane == 0) z[node] = s;
}

// ----------------------------------------- layer2 aggregation (scalar feature)
__global__ __launch_bounds__(256) void k_self2(const float* __restrict__ z,
                                               const float* __restrict__ dinv,
                                               const float* __restrict__ b2,
                                               float* __restrict__ out) {
    int i = blockIdx.x * 256 + threadIdx.x;
    if (i < N_NODES) {
        float n = dinv[i];
        out[i] = z[i] * n * n + b2[0];
    }
}

__global__ __launch_bounds__(256) void k_edge2(const int* __restrict__ src,
                                               const int* __restrict__ dst,
                                               const float* __restrict__ z,
                                               const float* __restrict__ dinv,
                                               float* __restrict__ out) {
    int e = blockIdx.x * 256 + threadIdx.x;
    if (e < N_EDGES) {
        int s = src[e], d = dst[e];
        atomicAdd(&out[d], z[s] * dinv[s] * dinv[d]);
    }
}

// ---------------------------------------------------------------------- launch
extern "C" void kernel_launch(void* const* d_in, const int* in_sizes, int n_in,
                              void* d_out, int out_size, void* d_ws, size_t ws_size,
                              hipStream_t stream) {
    const float* x   = (const float*)d_in[0];
    const float* W1  = (const float*)d_in[1];
    const float* b1  = (const float*)d_in[2];
    const float* W2  = (const float*)d_in[3];
    const float* b2  = (const float*)d_in[4];
    const int*   ei  = (const int*)d_in[5];
    const int*   src = ei;                 // edge_index[0]
    const int*   dst = ei + N_EDGES;       // edge_index[1]
    float*       out = (float*)d_out;

    char* ws = (char*)d_ws;
    const size_t NPAD = 409600;            // 100000*4 rounded up, keeps 4KB alignment
    float* deg  = (float*)(ws);
    float* dinv = (float*)(ws + NPAD);
    float* h    = (float*)(ws + 2 * NPAD);                               // 25.6 MB
    float* agg1 = (float*)(ws + 2 * NPAD + (size_t)N_NODES * H_F * 4);   // 25.6 MB
    float* z    = deg;                     // deg dead after k_dinv; reuse

    const int nb_nodes = (N_NODES + 255) / 256;              // 391
    const int nb_edges = (N_EDGES + 255) / 256;              // 6250

    k_init_deg<<<nb_nodes, 256, 0, stream>>>(deg);
    k_edge_deg<<<nb_edges, 256, 0, stream>>>(dst, deg);
    k_dinv    <<<nb_nodes, 256, 0, stream>>>(deg, dinv);

    k_gemm1   <<<(N_NODES + 127) / 128, 256, 0, stream>>>(x, W1, h);     // 782 blocks

    k_self1   <<<(N_NODES * 16) / 256, 256, 0, stream>>>(h, dinv, agg1); // 6250
    k_edge1   <<<(N_EDGES * 16) / 256, 256, 0, stream>>>(src, dst, h, dinv, agg1); // 100000

    k_dot     <<<(N_NODES * 32) / 256, 256, 0, stream>>>(agg1, b1, W2, z); // 12500

    k_self2   <<<nb_nodes, 256, 0, stream>>>(z, dinv, b2, out);
    k_edge2   <<<nb_edges, 256, 0, stream>>>(src, dst, z, dinv, out);
}